// GAT_5909874999495
// MI455X (gfx1250) — compile-verified
//
#include <hip/hip_runtime.h>
#include <hip/hip_bf16.h>
#include <math.h>

// ---------------------------------------------------------------------------
// GAT encoder (4 layers) + sigmoid(z z^T) decoder for MI455X (gfx1250).
//  - Dense math: V_WMMA_F32_16X16X4_F32 (native fp32 WMMA, wave32)
//  - GEMM staging: GLOBAL_LOAD_ASYNC_TO_LDS_B128 (ASYNCcnt) with double-
//    buffered LDS so next-slab loads overlap current-slab WMMA
//  - Edge softmax/aggregation: CSR-by-dst + one-wave-per-node online softmax
//    (no float atomics; BN+ReLU+L2-normalize fused into the epilogue)
// ---------------------------------------------------------------------------

#define HF        256      // hidden width (H)
#define GAT_SLOPE 0.2f
#define EPS_BN    1e-5f

typedef __attribute__((ext_vector_type(2))) float v2f;
typedef __attribute__((ext_vector_type(8))) float v8f;
typedef __attribute__((__vector_size__(16))) int i32x4;

#if defined(__gfx1250__) && __has_builtin(__builtin_amdgcn_global_load_async_to_lds_b128)
#define ASYNC_CP 1
#else
#define ASYNC_CP 0
#endif

#if ASYNC_CP
// builtin signature (from hipcc diagnostic): arg0 = i32x4* in global AS,
// arg1 = i32x4* in LDS AS, then two int immediates (offset, cpol)
#define GAS1(p) ((__attribute__((address_space(1))) i32x4*)(p))
#define LAS3(p) ((__attribute__((address_space(3))) i32x4*)(p))
__device__ __forceinline__ void async_wait0() {
#if __has_builtin(__builtin_amdgcn_s_wait_asynccnt)
  __builtin_amdgcn_s_wait_asynccnt(0);
#else
  asm volatile("s_wait_asynccnt 0" ::: "memory");
#endif
}
#endif

// ---------------------------------------------------------------------------
// C = A(MxK) * B, f32 WMMA. TRANS_B=false: B is [K,N] row-major.
// TRANS_B=true : B is [N,K] row-major (computes A * B^T).  EPI: 0=none,1=sigmoid
// Block: 128 threads (4 waves), tile 64x64, K double-buffered 16 at a time.
// Wave w owns rows [16w,16w+16) x 64 cols (4 x v8f accumulators).
// TRANS_B B-tile is staged ROW-major ([64][17]) and fragments are read down
// the padded columns -- the B fragment (fixed N per lane, K across regs) has
// the same access pattern as the A fragment, so no LDS transpose is needed.
// M,N multiples of 64; K multiple of 16 (true for all call sites here).
// ---------------------------------------------------------------------------
template<bool TRANS_B, int EPI>
__global__ __launch_bounds__(128) void gemm_f32_wmma(
    const float* __restrict__ A, const float* __restrict__ B,
    float* __restrict__ C, int M, int N, int K)
{
  constexpr int BSZ = TRANS_B ? 64 * 17 : 16 * 65;
  __shared__ float As[2][64][17];   // pad 17 (coprime w/ 64 banks)
  __shared__ float Bsm[2][BSZ];     // trans: [64][17] row-major; else [16][65]

  const int tid  = threadIdx.x;
  const int lane = tid & 31;
  const int wave = tid >> 5;     // 0..3
  const int half = lane >> 4;    // 0/1 half-wave
  const int l16  = lane & 15;
  const int n0   = blockIdx.x * 64;
  const int m0   = blockIdx.y * 64;

  v8f c0 = {}, c1 = {}, c2 = {}, c3 = {};

  // ---- stage one 16-wide K-slab of A and B into LDS buffer `buf` ----
  auto stage = [&](int buf, int kb) {
#if ASYNC_CP
    #pragma unroll
    for (int c = tid; c < 256; c += 128) {        // A: 64 rows x 16 f32
      const int row = c >> 2, cw = (c & 3) * 4;
      __builtin_amdgcn_global_load_async_to_lds_b128(
          GAS1(A + (size_t)(m0 + row) * K + kb + cw),
          LAS3(&As[buf][row][cw]), 0, 0);
    }
    if (!TRANS_B) {
      #pragma unroll
      for (int c = tid; c < 256; c += 128) {      // B: 16 rows x 64 f32
        const int row = c >> 4, cw = (c & 15) * 4;
        __builtin_amdgcn_global_load_async_to_lds_b128(
            GAS1(B + (size_t)(kb + row) * N + n0 + cw),
            LAS3(&Bsm[buf][row * 65 + cw]), 0, 0);
      }
    } else {
      #pragma unroll
      for (int c = tid; c < 256; c += 128) {      // B^T: 64 z-rows x 16 f32
        const int row = c >> 2, cw = (c & 3) * 4;
        __builtin_amdgcn_global_load_async_to_lds_b128(
            GAS1(B + (size_t)(n0 + row) * K + kb + cw),
            LAS3(&Bsm[buf][row * 17 + cw]), 0, 0);
      }
    }
#else
    {   // fallback: register staging (8 contiguous f32 per thread per tile)
      const int row = tid >> 1, col = (tid & 1) * 8;
      const float* src = A + (size_t)(m0 + row) * K + kb + col;
      #pragma unroll
      for (int j = 0; j < 8; ++j) As[buf][row][col + j] = src[j];
    }
    if (!TRANS_B) {
      const int row = tid >> 3, col = (tid & 7) * 8;
      const float* src = B + (size_t)(kb + row) * N + n0 + col;
      #pragma unroll
      for (int j = 0; j < 8; ++j) Bsm[buf][row * 65 + col + j] = src[j];
    } else {
      const int row = tid >> 1, col = (tid & 1) * 8;
      const float* src = B + (size_t)(n0 + row) * K + kb + col;
      #pragma unroll
      for (int j = 0; j < 8; ++j) Bsm[buf][row * 17 + col + j] = src[j];
    }
#endif
  };

  stage(0, 0);
#if ASYNC_CP
  async_wait0();
#endif
  __syncthreads();

  int buf = 0;
  for (int kb = 0; kb < K; kb += 16) {
    if (kb + 16 < K) stage(buf ^ 1, kb + 16);   // prefetch next slab (async)

    #pragma unroll
    for (int kk = 0; kk < 16; kk += 4) {
      // A 16x4: lanes 0-15 hold K=kk+0/+1, lanes 16-31 hold K=kk+2/+3
      const int r0 = kk + 2 * half;
      v2f a;
      a.x = As[buf][wave * 16 + l16][r0];
      a.y = As[buf][wave * 16 + l16][r0 + 1];
      // B 4x16: vgpr0 = K rows r0 (fixed col per lane), vgpr1 = r0+1
      v2f b0, b1, b2, b3;
      if (!TRANS_B) {
        b0.x = Bsm[buf][r0 * 65 +  0 + l16]; b0.y = Bsm[buf][(r0 + 1) * 65 +  0 + l16];
        b1.x = Bsm[buf][r0 * 65 + 16 + l16]; b1.y = Bsm[buf][(r0 + 1) * 65 + 16 + l16];
        b2.x = Bsm[buf][r0 * 65 + 32 + l16]; b2.y = Bsm[buf][(r0 + 1) * 65 + 32 + l16];
        b3.x = Bsm[buf][r0 * 65 + 48 + l16]; b3.y = Bsm[buf][(r0 + 1) * 65 + 48 + l16];
      } else {
        b0.x = Bsm[buf][( 0 + l16) * 17 + r0]; b0.y = Bsm[buf][( 0 + l16) * 17 + r0 + 1];
        b1.x = Bsm[buf][(16 + l16) * 17 + r0]; b1.y = Bsm[buf][(16 + l16) * 17 + r0 + 1];
        b2.x = Bsm[buf][(32 + l16) * 17 + r0]; b2.y = Bsm[buf][(32 + l16) * 17 + r0 + 1];
        b3.x = Bsm[buf][(48 + l16) * 17 + r0]; b3.y = Bsm[buf][(48 + l16) * 17 + r0 + 1];
      }
      c0 = __builtin_amdgcn_wmma_f32_16x16x4_f32(false, a, false, b0, (short)0, c0, false, false);
      c1 = __builtin_amdgcn_wmma_f32_16x16x4_f32(false, a, false, b1, (short)0, c1, false, false);
      c2 = __builtin_amdgcn_wmma_f32_16x16x4_f32(false, a, false, b2, (short)0, c2, false, false);
      c3 = __builtin_amdgcn_wmma_f32_16x16x4_f32(false, a, false, b3, (short)0, c3, false, false);
    }

#if ASYNC_CP
    async_wait0();          // next-slab async loads landed in LDS
#endif
    __syncthreads();        // all waves done reading `buf`, next buffer ready
    buf ^= 1;
  }

  // ---- store: C vgpr r = row (r + 8*half), lane l16 = column ----
  const int row_base = m0 + wave * 16 + half * 8;
  #pragma unroll
  for (int r = 0; r < 8; ++r) {
    size_t off = (size_t)(row_base + r) * N + n0 + l16;
    float v0 = c0[r], v1 = c1[r], v2 = c2[r], v3 = c3[r];
    if (EPI == 1) {
      v0 = 1.f / (1.f + __expf(-v0));
      v1 = 1.f / (1.f + __expf(-v1));
      v2 = 1.f / (1.f + __expf(-v2));
      v3 = 1.f / (1.f + __expf(-v3));
    }
    C[off] = v0; C[off + 16] = v1; C[off + 32] = v2; C[off + 48] = v3;
  }
}

// ---------------------------------------------------------------------------
// es[i] = h[i,:].att_src ; ed[i] = h[i,:].att_dst   (one wave per row)
// ---------------------------------------------------------------------------
__global__ __launch_bounds__(256) void row_dots(
    const float* __restrict__ h, const float* __restrict__ a_s,
    const float* __restrict__ a_d, float* __restrict__ es,
    float* __restrict__ ed, int N)
{
  const int gw   = (blockIdx.x * 256 + threadIdx.x) >> 5;
  const int lane = threadIdx.x & 31;
  if (gw >= N) return;
  const float* row = h + (size_t)gw * HF;
  float s = 0.f, d = 0.f;
  #pragma unroll
  for (int j = 0; j < 8; ++j) {
    const float v = row[lane + 32 * j];
    s += v * a_s[lane + 32 * j];
    d += v * a_d[lane + 32 * j];
  }
  #pragma unroll
  for (int off = 16; off > 0; off >>= 1) {
    s += __shfl_down(s, off, 32);
    d += __shfl_down(d, off, 32);
  }
  if (lane == 0) { es[gw] = s; ed[gw] = d; }
}

// ---------------------------------------------------------------------------
// Per-destination online softmax + aggregation + bias + BN + ReLU.
// One wave per destination node; 256 features = 8 floats/lane.
// LAST=true additionally L2-normalizes the row (final z).
// ---------------------------------------------------------------------------
template<bool LAST>
__global__ __launch_bounds__(256) void gat_aggregate(
    const float* __restrict__ h,        // [N,HF] = W*x (pre-attention)
    const int* __restrict__ row_ptr, const int* __restrict__ csr_src,
    const float* __restrict__ es, const float* __restrict__ ed,
    const float* __restrict__ bias,
    const float* __restrict__ bn_g, const float* __restrict__ bn_b,
    const float* __restrict__ bn_m, const float* __restrict__ bn_v,
    float* __restrict__ out, int N)
{
  const int wid  = (blockIdx.x * 256 + threadIdx.x) >> 5;
  const int lane = threadIdx.x & 31;
  if (wid >= N) return;

  const int   beg = row_ptr[wid];
  const int   end = row_ptr[wid + 1];      // >= beg+1 (self-loop guaranteed)
  const float edd = ed[wid];

  float m = -__builtin_inff();
  float l = 0.f;
  float acc[8];
  #pragma unroll
  for (int j = 0; j < 8; ++j) acc[j] = 0.f;

  for (int e = beg; e < end; ++e) {
    const int   s  = csr_src[e];
    const float v  = es[s] + edd;
    const float zl = v > 0.f ? v : GAT_SLOPE * v;   // leaky_relu logit
    const float mn    = fmaxf(m, zl);
    const float scale = __expf(m - mn);             // exp(-inf)=0 on 1st iter
    const float w     = __expf(zl - mn);
    l = l * scale + w;
    const float* hr = h + (size_t)s * HF;
    #pragma unroll
    for (int j = 0; j < 8; ++j)
      acc[j] = acc[j] * scale + w * hr[lane + 32 * j];
    m = mn;
  }

  const float inv = 1.f / l;
  float y[8];
  #pragma unroll
  for (int j = 0; j < 8; ++j) {
    const int c = lane + 32 * j;
    float o = acc[j] * inv + bias[c];
    o = (o - bn_m[c]) * rsqrtf(bn_v[c] + EPS_BN) * bn_g[c] + bn_b[c];
    y[j] = fmaxf(o, 0.f);
  }

  if (LAST) {
    float ss = 0.f;
    #pragma unroll
    for (int j = 0; j < 8; ++j) ss += y[j] * y[j];
    #pragma unroll
    for (int off = 16; off > 0; off >>= 1) ss += __shfl_xor(ss, off, 32);
    const float nrm = sqrtf(ss);
    const float si  = 1.f / fmaxf(nrm, 1e-12f);
    #pragma unroll
    for (int j = 0; j < 8; ++j) out[(size_t)wid * HF + lane + 32 * j] = y[j] * si;
  } else {
    #pragma unroll
    for (int j = 0; j < 8; ++j) out[(size_t)wid * HF + lane + 32 * j] = y[j];
  }
}

// ---------------------------------------------------------------------------
// CSR-by-destination build (run every launch; cheap, ~270K edges)
// ---------------------------------------------------------------------------
__global__ void init_i32(int* __restrict__ p, int v, int n) {
  const int i = blockIdx.x * blockDim.x + threadIdx.x;
  if (i < n) p[i] = v;
}

__global__ void count_edges(const long long* __restrict__ ei, int E, int N,
                            int* __restrict__ cnt) {
  const int i = blockIdx.x * blockDim.x + threadIdx.x;
  if (i >= E + N) return;
  const int d = (i < E) ? (int)ei[(size_t)E + i] : (i - E);  // dst row / self loop
  atomicAdd(&cnt[d], 1);
}

// exclusive scan of 8192 counts with one 1024-thread block (8 elems/thread)
__global__ __launch_bounds__(1024) void scan_counts(
    const int* __restrict__ cnt, int* __restrict__ row_ptr,
    int* __restrict__ cursor, int n)
{
  __shared__ int sums[1024];
  const int t    = threadIdx.x;
  const int base = t * 8;
  int local[8], s = 0;
  #pragma unroll
  for (int j = 0; j < 8; ++j) { local[j] = cnt[base + j]; s += local[j]; }
  sums[t] = s;
  __syncthreads();
  for (int off = 1; off < 1024; off <<= 1) {   // Hillis-Steele inclusive scan
    const int v = (t >= off) ? sums[t - off] : 0;
    __syncthreads();
    sums[t] += v;
    __syncthreads();
  }
  int run = (t == 0) ? 0 : sums[t - 1];
  #pragma unroll
  for (int j = 0; j < 8; ++j) {
    row_ptr[base + j] = run;
    cursor[base + j]  = run;
    run += local[j];
  }
  if (t == 1023) row_ptr[n] = run;   // total = E + N
}

__global__ void fill_csr(const long long* __restrict__ ei, int E, int N,
                         int* __restrict__ cursor, int* __restrict__ csr_src) {
  const int i = blockIdx.x * blockDim.x + threadIdx.x;
  if (i >= E + N) return;
  int s, d;
  if (i < E) { s = (int)ei[i]; d = (int)ei[(size_t)E + i]; }
  else       { s = d = i - E; }
  const int pos = atomicAdd(&cursor[d], 1);
  csr_src[pos] = s;
}

// ---------------------------------------------------------------------------
extern "C" void kernel_launch(void* const* d_in, const int* in_sizes, int n_in,
                              void* d_out, int out_size, void* d_ws, size_t ws_size,
                              hipStream_t stream) {
  (void)n_in; (void)out_size; (void)ws_size;

  const float*     x        = (const float*)d_in[0];
  const long long* ei       = (const long long*)d_in[1];
  const float*     W0       = (const float*)d_in[2];
  const float*     att_s0   = (const float*)d_in[3];
  const float*     att_d0   = (const float*)d_in[4];
  const float*     bias0    = (const float*)d_in[5];
  const float*     W        = (const float*)d_in[6];
  const float*     att_s    = (const float*)d_in[7];
  const float*     att_d    = (const float*)d_in[8];
  const float*     bias     = (const float*)d_in[9];
  const float*     bn_g     = (const float*)d_in[10];
  const float*     bn_b     = (const float*)d_in[11];
  const float*     bn_m     = (const float*)d_in[12];
  const float*     bn_v     = (const float*)d_in[13];

  const int N   = 8192;
  const int E   = in_sizes[1] / 2;      // 262144
  const int FIN = 512;
  const int H   = HF;                   // 256
  const int TOT = E + N;

  // workspace layout
  float* h_pre  = (float*)d_ws;               // [N,H]  W*x (pre-attention)
  float* h_post = h_pre  + (size_t)N * H;     // [N,H]  layer output
  float* es     = h_post + (size_t)N * H;     // [N]
  float* ed     = es + N;                     // [N]
  int*   cnt     = (int*)(ed + N);            // [N]
  int*   row_ptr = cnt + N;                   // [N+1]
  int*   cursor  = row_ptr + (N + 1);         // [N]
  int*   csr_src = cursor + N;                // [E+N]

  float* A_pred = (float*)d_out;                       // [N,N]
  float* z_out  = A_pred + (size_t)N * N;              // [N,H]

  // ---- CSR build (deterministic up to intra-bucket summation order) ----
  init_i32    <<<(N + 255) / 256,   256, 0, stream>>>(cnt, 0, N);
  count_edges <<<(TOT + 255) / 256, 256, 0, stream>>>(ei, E, N, cnt);
  scan_counts <<<1, 1024, 0, stream>>>(cnt, row_ptr, cursor, N);
  fill_csr    <<<(TOT + 255) / 256, 256, 0, stream>>>(ei, E, N, cursor, csr_src);

  const dim3 gemm_blk(128);
  const dim3 gemm_grid_h(H / 64, N / 64);   // layer GEMMs: [N,H]

  // ---- layer 0: h_pre = x @ W0 ; attention ; aggregate+BN+ReLU ----
  gemm_f32_wmma<false, 0><<<gemm_grid_h, gemm_blk, 0, stream>>>(x, W0, h_pre, N, H, FIN);
  row_dots<<<N / 8, 256, 0, stream>>>(h_pre, att_s0, att_d0, es, ed, N);
  gat_aggregate<false><<<N / 8, 256, 0, stream>>>(
      h_pre, row_ptr, csr_src, es, ed, bias0,
      bn_g + 0 * H, bn_b + 0 * H, bn_m + 0 * H, bn_v + 0 * H, h_post, N);

  // ---- layers 1..3 ----
  for (int i = 0; i < 3; ++i) {
    gemm_f32_wmma<false, 0><<<gemm_grid_h, gemm_blk, 0, stream>>>(
        h_post, W + (size_t)i * H * H, h_pre, N, H, H);
    row_dots<<<N / 8, 256, 0, stream>>>(h_pre, att_s + (size_t)i * H,
                                        att_d + (size_t)i * H, es, ed, N);
    if (i < 2) {
      gat_aggregate<false><<<N / 8, 256, 0, stream>>>(
          h_pre, row_ptr, csr_src, es, ed, bias + (size_t)i * H,
          bn_g + (size_t)(i + 1) * H, bn_b + (size_t)(i + 1) * H,
          bn_m + (size_t)(i + 1) * H, bn_v + (size_t)(i + 1) * H, h_post, N);
    } else {   // final layer: fused L2-normalize, write z straight into d_out
      gat_aggregate<true><<<N / 8, 256, 0, stream>>>(
          h_pre, row_ptr, csr_src, es, ed, bias + (size_t)i * H,
          bn_g + (size_t)(i + 1) * H, bn_b + (size_t)(i + 1) * H,
          bn_m + (size_t)(i + 1) * H, bn_v + (size_t)(i + 1) * H, z_out, N);
    }
  }

  // ---- decoder: A_pred = sigmoid(z @ z^T), 34.4 GFLOP fp32 WMMA ----
  gemm_f32_wmma<true, 1><<<dim3(N / 64, N / 64), gemm_blk, 0, stream>>>(
      z_out, z_out, A_pred, N, N, H);
}